// FCOS_45784351375861
// MI455X (gfx1250) — compile-verified
//
#include <hip/hip_runtime.h>
#include <hip/hip_bf16.h>
#include <math.h>
#include <stdint.h>

#define IMGF      2048.0f
#define NCLS      80
#define TOPK      1000
#define NCAND     4256
#define NWORDS    133          // ceil(4256/32)
#define NPAD      8192
#define CONF      0.05f
#define NMS_T     0.6f
#define SC_OFF    (NCAND*4)            // 17024
#define LB_OFF    (NCAND*4 + NCAND)    // 21280
#define KP_OFF    (NCAND*4 + 2*NCAND)  // 25536

__constant__ int dLVL_M[5]     = {65536,16384,4096,1024,256};
__constant__ int dLVL_BASE[5]  = {0,65536,81920,86016,87040};
__constant__ int dLVL_H[5]     = {256,128,64,32,16};
__constant__ int dLVL_LOG2H[5] = {8,7,6,5,4};
__constant__ int dLVL_S[5]     = {8,16,32,64,128};
__constant__ int dCAND_BASE[5] = {0,1000,2000,3000,4000};

__device__ __forceinline__ float sigm(float x){ return 1.0f/(1.0f+expf(-x)); }

// ---------------------------------------------------------------------------
// K1: per-anchor max/argmax over 80 classes + fused score.
// Stages a 32-anchor x 80-class tile (10240 B) into LDS via CDNA5 async
// global->LDS B128 loads, then one wave32 per anchor does a shuffle reduction.
// ---------------------------------------------------------------------------
__global__ void k_scores(const float* __restrict__ cls, const float* __restrict__ ctn,
                         float* __restrict__ scoresAll, int* __restrict__ labelsAll, int base)
{
    __shared__ float tile[32*NCLS];                 // 10240 bytes
    const int tid = threadIdx.x;
    const int a0  = blockIdx.x * 32;

    const char* gsrc = (const char*)cls + (size_t)a0 * NCLS * 4;
    unsigned ldsBase = (unsigned)(uintptr_t)(&tile[0]);  // low 32 bits of generic LDS ptr = LDS offset
    for (int ch = tid; ch < (32*NCLS*4)/16; ch += blockDim.x) {
        unsigned    loff = ldsBase + (unsigned)ch*16u;
        const char* gp   = gsrc + (size_t)ch*16;
        asm volatile("global_load_async_to_lds_b128 %0, %1, off"
                     :: "v"(loff), "v"(gp) : "memory");
    }
    asm volatile("s_wait_asynccnt 0" ::: "memory");
    __syncthreads();

    const int wave = tid >> 5, lane = tid & 31;
    for (int q = 0; q < 4; ++q) {
        const int a = wave*4 + q;                   // anchor within tile (8 waves * 4)
        const float* row = &tile[a*NCLS];
        float bv = row[lane]; int bi = lane;        // classes: lane, lane+32, lane+64(<80)
        float v1 = row[lane+32]; if (v1 > bv){ bv=v1; bi=lane+32; }
        if (lane < 16){ float v2 = row[lane+64]; if (v2 > bv){ bv=v2; bi=lane+64; } }
        for (int off = 16; off >= 1; off >>= 1) {
            float ov = __shfl_xor(bv, off, 32);
            int   oi = __shfl_xor(bi, off, 32);
            if (ov > bv || (ov == bv && oi < bi)) { bv = ov; bi = oi; }
        }
        if (lane == 0) {
            int idx = a0 + a;
            float s = sqrtf(sigm(bv) * sigm(ctn[idx]));  // sqrt(sig(max cls)*sig(ctn))
            scoresAll[base+idx] = s;
            labelsAll[base+idx] = bi;
        }
    }
}

// ---------------------------------------------------------------------------
// K2: per-level (0..3) radix-select of the 1000th largest score (positive
// floats -> bit-monotone), then compact key>=T as packed (bits<<32)|~idx.
// ---------------------------------------------------------------------------
__global__ void k_topk_select(const float* __restrict__ scoresAll,
                              unsigned long long* __restrict__ keysLvl,
                              int* __restrict__ countC)
{
    __shared__ unsigned hist[256];
    __shared__ unsigned sPrefix, sRemaining, sCnt;
    const int l = blockIdx.x, tid = threadIdx.x, nt = blockDim.x;
    const int m = dLVL_M[l], base = dLVL_BASE[l];
    if (tid == 0){ sPrefix = 0u; sRemaining = TOPK; }
    __syncthreads();
    for (int p = 0; p < 4; ++p) {
        for (int b = tid; b < 256; b += nt) hist[b] = 0u;
        __syncthreads();
        const unsigned prefix = sPrefix;
        const int shift = 24 - 8*p;
        for (int i = tid; i < m; i += nt) {
            unsigned k = __float_as_uint(scoresAll[base+i]);
            if (p == 0 || (k >> (shift+8)) == prefix)
                atomicAdd(&hist[(k >> shift) & 255u], 1u);
        }
        __syncthreads();
        if (tid == 0) {
            unsigned cum = 0, rem = sRemaining;
            for (int b = 255; b >= 0; --b) {
                unsigned h = hist[b];
                if (cum + h >= rem) { sPrefix = (prefix<<8) | (unsigned)b; sRemaining = rem - cum; break; }
                cum += h;
            }
        }
        __syncthreads();
    }
    const unsigned T = sPrefix;
    if (tid == 0) sCnt = 0u;
    __syncthreads();
    for (int i = tid; i < m; i += nt) {
        unsigned k = __float_as_uint(scoresAll[base+i]);
        if (k >= T) {
            unsigned pos = atomicAdd(&sCnt, 1u);
            if (pos < 2048u)
                keysLvl[l*2048 + pos] = ((unsigned long long)k << 32) | (unsigned)(~(unsigned)i);
        }
    }
    __syncthreads();
    if (tid == 0) countC[l] = (int)(sCnt < 2048u ? sCnt : 2048u);
}

// ---------------------------------------------------------------------------
// K3: per-level sorted top-k emit + box decode. LDS bitonic sort (desc by
// score bits, asc by index on ties == jax top_k). Level 4 passes through.
// ---------------------------------------------------------------------------
__global__ void k_emit(const float* __restrict__ scoresAll, const int* __restrict__ labelsAll,
                       const unsigned long long* __restrict__ keysLvl, const int* __restrict__ countC,
                       const float* r0, const float* r1, const float* r2, const float* r3, const float* r4,
                       const float* __restrict__ scales,
                       float* __restrict__ candScore, int* __restrict__ candLabel,
                       float* __restrict__ candBox, float* __restrict__ out)
{
    __shared__ unsigned long long sk[2048];
    const int l = blockIdx.x, tid = threadIdx.x;
    const float* regs[5] = {r0,r1,r2,r3,r4};
    const float* regp = regs[l];
    const float scale  = scales[l];
    const float stride = (float)dLVL_S[l];
    const int h = dLVL_H[l], log2h = dLVL_LOG2H[l];
    const int lvlBase = dLVL_BASE[l], candBase = dCAND_BASE[l];
    const int cnt = (l < 4) ? TOPK : 256;

    if (l < 4) {
        const int C = countC[l];
        for (int j = tid; j < 2048; j += blockDim.x)
            sk[j] = (j < C) ? keysLvl[l*2048 + j] : 0ull;
        __syncthreads();
        for (unsigned k = 2; k <= 2048; k <<= 1) {
            for (unsigned jj = k >> 1; jj > 0; jj >>= 1) {
                for (unsigned t = tid; t < 2048; t += blockDim.x) {
                    unsigned ixj = t ^ jj;
                    if (ixj > t) {
                        bool desc = ((t & k) == 0);
                        unsigned long long a = sk[t], b = sk[ixj];
                        bool sw = desc ? (a < b) : (a > b);
                        if (sw) { sk[t] = b; sk[ixj] = a; }
                    }
                }
                __syncthreads();
            }
        }
    }
    for (int j = tid; j < cnt; j += blockDim.x) {
        int idx; float score; int label;
        if (l < 4) {
            unsigned long long key = sk[j];
            idx   = (int)(~(unsigned)(key & 0xFFFFFFFFull));
            score = __uint_as_float((unsigned)(key >> 32));
            label = labelsAll[lvlBase + idx];
        } else {
            idx   = j;
            score = scoresAll[lvlBase + idx];
            label = labelsAll[lvlBase + idx];
        }
        float ax = ((float)(idx & (h-1)) + 0.5f) * stride;
        float ay = ((float)(idx >> log2h) + 0.5f) * stride;
        float e0 = fmaxf(regp[idx*4+0]*scale, 0.f) * stride;
        float e1 = fmaxf(regp[idx*4+1]*scale, 0.f) * stride;
        float e2 = fmaxf(regp[idx*4+2]*scale, 0.f) * stride;
        float e3 = fmaxf(regp[idx*4+3]*scale, 0.f) * stride;
        float x1 = ax - e0, y1 = ay - e1, x2 = ax + e2, y2 = ay + e3;
        int pos = candBase + j;
        candScore[pos] = score; candLabel[pos] = label;
        candBox[pos*4+0] = x1; candBox[pos*4+1] = y1;
        candBox[pos*4+2] = x2; candBox[pos*4+3] = y2;
        out[pos*4+0] = fminf(fmaxf(x1/IMGF, 0.f), 1.f);
        out[pos*4+1] = fminf(fmaxf(y1/IMGF, 0.f), 1.f);
        out[pos*4+2] = fminf(fmaxf(x2/IMGF, 0.f), 1.f);
        out[pos*4+3] = fminf(fmaxf(y2/IMGF, 0.f), 1.f);
        out[SC_OFF + pos] = score;
        out[LB_OFF + pos] = (float)label;
    }
}

// ---------------------------------------------------------------------------
// K4: global sort of all 4256 candidates (stable argsort(-scores) semantics),
// build class-offset boxes, areas, valid mask in sorted order.
// ---------------------------------------------------------------------------
__global__ void k_sortall(const float* __restrict__ candScore, const int* __restrict__ candLabel,
                          const float* __restrict__ candBox,
                          unsigned long long* __restrict__ keys8, int* __restrict__ sorder,
                          float* __restrict__ sbox, float* __restrict__ sarea, int* __restrict__ svalid)
{
    const int tid = threadIdx.x, nt = blockDim.x;
    for (int i = tid; i < NPAD; i += nt)
        keys8[i] = (i < NCAND)
            ? (((unsigned long long)__float_as_uint(candScore[i]) << 32) | (unsigned)(~(unsigned)i))
            : 0ull;
    __syncthreads();
    for (unsigned k = 2; k <= NPAD; k <<= 1) {
        for (unsigned jj = k >> 1; jj > 0; jj >>= 1) {
            for (unsigned t = tid; t < NPAD; t += nt) {
                unsigned ixj = t ^ jj;
                if (ixj > t) {
                    bool desc = ((t & k) == 0);
                    unsigned long long a = keys8[t], b = keys8[ixj];
                    bool sw = desc ? (a < b) : (a > b);
                    if (sw) { keys8[t] = b; keys8[ixj] = a; }
                }
            }
            __syncthreads();
        }
    }
    for (int i = tid; i < NCAND; i += nt) {
        unsigned long long key = keys8[i];
        int pos = (int)(~(unsigned)(key & 0xFFFFFFFFull));
        sorder[i] = pos;
        float sc  = __uint_as_float((unsigned)(key >> 32));
        float off = (float)candLabel[pos] * (4.0f * IMGF);
        float x1 = candBox[pos*4+0] + off, y1 = candBox[pos*4+1] + off;
        float x2 = candBox[pos*4+2] + off, y2 = candBox[pos*4+3] + off;
        sbox[i*4+0] = x1; sbox[i*4+1] = y1; sbox[i*4+2] = x2; sbox[i*4+3] = y2;
        sarea[i]  = (x2 - x1) * (y2 - y1);
        svalid[i] = (sc >= CONF) ? 1 : 0;
    }
}

// ---------------------------------------------------------------------------
// K5: suppression bit-matrix (4256 rows x 133 words), faithful clamps.
// ---------------------------------------------------------------------------
__global__ void k_suppr(const float* __restrict__ sbox, const float* __restrict__ sarea,
                        unsigned* __restrict__ suppr)
{
    __shared__ float bi[5];
    const int i = blockIdx.x, t = threadIdx.x;
    if (t == 0) { bi[0]=sbox[i*4+0]; bi[1]=sbox[i*4+1]; bi[2]=sbox[i*4+2]; bi[3]=sbox[i*4+3]; bi[4]=sarea[i]; }
    __syncthreads();
    if (t < NWORDS) {
        unsigned w = 0;
        for (int b = 0; b < 32; ++b) {
            int j = t*32 + b;
            if (j < NCAND && j > i) {
                float xx1 = fmaxf(bi[0], sbox[j*4+0]);
                float yy1 = fmaxf(bi[1], sbox[j*4+1]);
                float xx2 = fminf(bi[2], sbox[j*4+2]);
                float yy2 = fminf(bi[3], sbox[j*4+3]);
                float inter = fmaxf(1e-10f, xx2-xx1) * fmaxf(1e-10f, yy2-yy1);
                float ovr   = inter / (bi[4] + sarea[j] - inter + 1e-14f);
                if (ovr > NMS_T) w |= (1u << b);
            }
        }
        suppr[(size_t)i*NWORDS + t] = w;
    }
}

// ---------------------------------------------------------------------------
// K6: serial greedy NMS scan over the bitmask; scatter keep to cand order.
// ---------------------------------------------------------------------------
__global__ void k_nms(const unsigned* __restrict__ suppr, const int* __restrict__ svalid,
                      const int* __restrict__ sorder, float* __restrict__ out)
{
    __shared__ unsigned keepW[NWORDS];
    __shared__ int sFlag;
    const int t = threadIdx.x;
    if (t < NWORDS) {
        unsigned w = 0;
        for (int b = 0; b < 32; ++b) { int j = t*32+b; if (j < NCAND && svalid[j]) w |= (1u << b); }
        keepW[t] = w;
    }
    __syncthreads();
    for (int i = 0; i < NCAND; ++i) {
        if (t == 0) sFlag = (int)((keepW[i>>5] >> (i & 31)) & 1u);
        __syncthreads();
        if (sFlag && t < NWORDS) keepW[t] &= ~suppr[(size_t)i*NWORDS + t];
        __syncthreads();
    }
    for (int i = t; i < NCAND; i += blockDim.x) {
        int pos = sorder[i];
        out[KP_OFF + pos] = ((keepW[i>>5] >> (i & 31)) & 1u) ? 1.0f : 0.0f;
    }
}

extern "C" void kernel_launch(void* const* d_in, const int* in_sizes, int n_in,
                              void* d_out, int out_size, void* d_ws, size_t ws_size,
                              hipStream_t stream)
{
    const float* cls[5]; const float* reg[5]; const float* ctn[5];
    for (int l = 0; l < 5; ++l) {
        cls[l] = (const float*)d_in[3*l+0];
        reg[l] = (const float*)d_in[3*l+1];
        ctn[l] = (const float*)d_in[3*l+2];
    }
    const float* scales = (const float*)d_in[15];

    char* ws = (char*)d_ws;
    size_t off = 0;
    auto alloc = [&](size_t bytes) -> void* { void* p = ws + off; off = (off + bytes + 255) & ~(size_t)255; return p; };
    float*              scoresAll = (float*)alloc(87296*4);
    int*                labelsAll = (int*)alloc(87296*4);
    unsigned long long* keysLvl   = (unsigned long long*)alloc(4*2048*8);
    int*                countC    = (int*)alloc(4*4);
    float*              candScore = (float*)alloc(NCAND*4);
    int*                candLabel = (int*)alloc(NCAND*4);
    float*              candBox   = (float*)alloc(NCAND*16);
    unsigned long long* keys8     = (unsigned long long*)alloc((size_t)NPAD*8);
    int*                sorder    = (int*)alloc(NCAND*4);
    float*              sbox      = (float*)alloc(NCAND*16);
    float*              sarea     = (float*)alloc(NCAND*4);
    int*                svalid    = (int*)alloc(NCAND*4);
    unsigned*           suppr     = (unsigned*)alloc((size_t)NCAND*NWORDS*4);

    static const int M[5]    = {65536,16384,4096,1024,256};
    static const int BASE[5] = {0,65536,81920,86016,87040};

    for (int l = 0; l < 5; ++l)
        k_scores<<<M[l]/32, 256, 0, stream>>>(cls[l], ctn[l], scoresAll, labelsAll, BASE[l]);
    k_topk_select<<<4, 256, 0, stream>>>(scoresAll, keysLvl, countC);
    k_emit<<<5, 256, 0, stream>>>(scoresAll, labelsAll, keysLvl, countC,
                                  reg[0], reg[1], reg[2], reg[3], reg[4], scales,
                                  candScore, candLabel, candBox, (float*)d_out);
    k_sortall<<<1, 1024, 0, stream>>>(candScore, candLabel, candBox, keys8, sorder, sbox, sarea, svalid);
    k_suppr<<<NCAND, 256, 0, stream>>>(sbox, sarea, suppr);
    k_nms<<<1, 256, 0, stream>>>(suppr, svalid, sorder, (float*)d_out);
}